// PINN_Difference_RNN_81166291960584
// MI455X (gfx1250) — compile-verified
//
#include <hip/hip_runtime.h>
#include <math.h>

// Problem constants
#define TLEN 4194304               // total steps (2^22)
#define LCH  64                    // chunk length (per scan leaf)
#define NCH  65536                 // number of chunks = TLEN / LCH
#define CPW  16                    // chunks per wave in pass 3
#define TILES 4                    // 16-step tiles per chunk (LCH/16)

typedef float v2f __attribute__((ext_vector_type(2)));
typedef float v8f __attribute__((ext_vector_type(8)));

// ---------------------------------------------------------------------------
// Kernel 1: per-chunk accumulated offset  S_k = sum_{i=0..63} A^{63-i} d_i
// via y <- A y + d  (d = bA+bB + W_B u).  LDS-staged, coalesced b64 loads.
// Each 256-thread block handles 256 consecutive chunks in 4 phases of 16 steps.
// ---------------------------------------------------------------------------
__global__ __launch_bounds__(256) void pinn_k1_chunksum(
    const float* __restrict__ u,
    const float* __restrict__ WA, const float* __restrict__ bA,
    const float* __restrict__ WB, const float* __restrict__ bB,
    float2* __restrict__ S) {
  __shared__ __align__(16) float su0[256 * 17];   // stride 17: no bank conflicts
  __shared__ __align__(16) float su1[256 * 17];
  int tid = threadIdx.x;
  int kbase = blockIdx.x * 256;
  float wa00 = WA[0], wa01 = WA[1], wa10 = WA[2], wa11 = WA[3];
  float wb00 = WB[0], wb01 = WB[1], wb10 = WB[2], wb11 = WB[3];
  float bs0 = bA[0] + bB[0], bs1 = bA[1] + bB[1];
  float s0 = 0.f, s1 = 0.f;
  for (int ph = 0; ph < 4; ++ph) {
    // stage 256 chunks x 16 steps with contiguous float2 loads
#pragma unroll
    for (int rr = 0; rr < 8; ++rr) {
      int idx = rr * 512 + tid * 2;          // 0..4094, even
      int c = idx >> 4, i = idx & 15;        // i even
      size_t tt = (size_t)(kbase + c) * LCH + ph * 16 + i;
      float2 a0 = *(const float2*)(u + tt);
      float2 a1 = *(const float2*)(u + (size_t)TLEN + tt);
      *(float2*)&su0[c * 17 + i] = a0;
      *(float2*)&su1[c * 17 + i] = a1;
    }
    __syncthreads();
#pragma unroll
    for (int i = 0; i < 16; ++i) {
      float uu0 = su0[tid * 17 + i];
      float uu1 = su1[tid * 17 + i];
      float d0 = fmaf(wb00, uu0, fmaf(wb01, uu1, bs0));
      float d1 = fmaf(wb10, uu0, fmaf(wb11, uu1, bs1));
      float n0 = fmaf(wa00, s0, fmaf(wa01, s1, d0));
      float n1 = fmaf(wa10, s0, fmaf(wa11, s1, d1));
      s0 = n0; s1 = n1;
    }
    __syncthreads();
  }
  S[kbase + tid] = make_float2(s0, s1);
}

// ---------------------------------------------------------------------------
// Kernel 2: 3-level scan over chunk boundaries: X_{k+1} = A^64 X_k + S_k
// ---------------------------------------------------------------------------
__device__ __forceinline__ void matsq(float& a, float& b, float& c, float& d) {
  float na = fmaf(a, a, b * c);
  float nb = fmaf(a, b, b * d);
  float nc = fmaf(c, a, d * c);
  float nd = fmaf(c, b, d * d);
  a = na; b = nb; c = nc; d = nd;
}

__global__ void pinn_k2_scan(const float* __restrict__ x0,
                             const float* __restrict__ WA,
                             const float2* __restrict__ S,
                             float2* __restrict__ X) {
  __shared__ float Q[256][2];
  __shared__ float Pst[256][2];
  int j = threadIdx.x;
  float a = WA[0], b = WA[1], c = WA[2], d = WA[3];
  for (int it = 0; it < 6; ++it) matsq(a, b, c, d);          // G = A^64
  const float2* Sj = S + (size_t)j * 256;
  float y0 = 0.f, y1 = 0.f;
  for (int i = 0; i < 256; ++i) {
    float2 s = Sj[i];
    float t0 = fmaf(a, y0, fmaf(b, y1, s.x));
    float t1 = fmaf(c, y0, fmaf(d, y1, s.y));
    y0 = t0; y1 = t1;
  }
  Q[j][0] = y0; Q[j][1] = y1;
  __syncthreads();
  if (j == 0) {
    float ha = a, hb = b, hc = c, hd = d;
    for (int it = 0; it < 8; ++it) matsq(ha, hb, hc, hd);    // H = G^256
    float p0 = x0[0], p1 = x0[1];
    for (int q = 0; q < 256; ++q) {
      Pst[q][0] = p0; Pst[q][1] = p1;
      float t0 = fmaf(ha, p0, fmaf(hb, p1, Q[q][0]));
      float t1 = fmaf(hc, p0, fmaf(hd, p1, Q[q][1]));
      p0 = t0; p1 = t1;
    }
  }
  __syncthreads();
  float p0 = Pst[j][0], p1 = Pst[j][1];
  float2* Xj = X + (size_t)j * 256;
  for (int i = 0; i < 256; ++i) {
    Xj[i] = make_float2(p0, p1);
    float2 s = Sj[i];
    float t0 = fmaf(a, p0, fmaf(b, p1, s.x));
    float t1 = fmaf(c, p0, fmaf(d, p1, s.y));
    p0 = t0; p1 = t1;
  }
}

// ---------------------------------------------------------------------------
// Kernel 3: WMMA state expansion + output.
// One wave per block; wave handles 16 chunks, 16 steps per tile.
// D(32x16) = Aop(32x32) * Bop(32x16): jj = 0..15 (e_0 = init, e_jj = d_{jj-1}).
// Row-block 0 (i=0..7) needs only K-slabs 0..3; block 1 needs slabs 0..7
// -> 12 x V_WMMA_F32_16X16X4_F32 per 256 steps, A operand constant.
// ---------------------------------------------------------------------------
__global__ __launch_bounds__(32) void pinn_k3_wmma(
    const float* __restrict__ u, const float* __restrict__ td,
    const float* __restrict__ WA, const float* __restrict__ bA,
    const float* __restrict__ WB, const float* __restrict__ bB,
    const float2* __restrict__ X, float2* __restrict__ out2) {
  __shared__ float Ppow[16][2][2];               // A^n, n=0..15
  __shared__ float sArr[17][16][2];              // states s_0..s_16 per chunk
  __shared__ __align__(16) float u0s[256];
  __shared__ __align__(16) float u1s[256];
  __shared__ __align__(16) float dts[256];

  int lane = threadIdx.x;
  int p = lane >> 4;        // lane-half -> jj parity
  int cidx = lane & 15;     // chunk column for B/C/D
  int chunkBase = blockIdx.x * CPW;

  float wa00 = WA[0], wa01 = WA[1], wa10 = WA[2], wa11 = WA[3];
  float wb00 = WB[0], wb01 = WB[1], wb10 = WB[2], wb11 = WB[3];
  float bs0 = bA[0] + bB[0], bs1 = bA[1] + bB[1];

  if (lane == 0) {                                // A^n table
    float c00 = 1.f, c01 = 0.f, c10 = 0.f, c11 = 1.f;
    for (int n = 0; n < 16; ++n) {
      Ppow[n][0][0] = c00; Ppow[n][0][1] = c01;
      Ppow[n][1][0] = c10; Ppow[n][1][1] = c11;
      float n00 = fmaf(wa00, c00, wa01 * c10);
      float n01 = fmaf(wa00, c01, wa01 * c11);
      float n10 = fmaf(wa10, c00, wa11 * c10);
      float n11 = fmaf(wa10, c01, wa11 * c11);
      c00 = n00; c01 = n01; c10 = n10; c11 = n11;
    }
  }
  if (lane < 16) {                                // initial chunk states
    float2 xi = X[chunkBase + lane];
    sArr[16][lane][0] = xi.x;
    sArr[16][lane][1] = xi.y;
  }
  __syncthreads();

  // Constant A operand: Av[b][s][v] = (jj<=i) ? A^{i-jj}[r][v] : 0,
  // jj = 2s+p, i = 8b + il, r = lane&1  (A-layout: VGPR v -> K=4s+v / +2).
  int r = lane & 1;
  int il = (lane & 15) >> 1;
  float Av0[4][2], Av1[8][2];
#pragma unroll
  for (int s = 0; s < 4; ++s) {
    int jj = 2 * s + p;
#pragma unroll
    for (int v = 0; v < 2; ++v)
      Av0[s][v] = (jj <= il) ? Ppow[il - jj][r][v] : 0.0f;
  }
#pragma unroll
  for (int s = 0; s < 8; ++s) {
    int jj = 2 * s + p;
    int i = 8 + il;
#pragma unroll
    for (int v = 0; v < 2; ++v)
      Av1[s][v] = (jj <= i) ? Ppow[i - jj][r][v] : 0.0f;
  }

  for (int tile = 0; tile < TILES; ++tile) {
    int tbase = tile * 16;
    // Stage u0/u1/dt (16 chunks x 16 steps) with contiguous float2 loads.
#pragma unroll
    for (int rr = 0; rr < 4; ++rr) {
      int idx = rr * 64 + lane * 2;             // even
      int c = idx >> 4, i = idx & 15;
      size_t tt = (size_t)(chunkBase + c) * LCH + tbase + i;
      float2 a0 = *(const float2*)(u + tt);
      float2 a1 = *(const float2*)(u + (size_t)TLEN + tt);
      float2 a2 = *(const float2*)(td + tt);
      *(float2*)&u0s[idx] = a0;
      *(float2*)&u1s[idx] = a1;
      *(float2*)&dts[idx] = a2;
    }
    __syncthreads();

    // Build B operand: Bop[(jj,cc), chunk]; e_0 = init, e_jj = d_{jj-1}.
    float Bv[8][2];
    if (p == 0) {                    // even jj: jj=0 init, jj=2..14 -> j odd
      Bv[0][0] = sArr[16][cidx][0];
      Bv[0][1] = sArr[16][cidx][1];
#pragma unroll
      for (int s = 1; s < 8; ++s) {
        int f = cidx * 16 + (2 * s - 1);
        float uu0 = u0s[f], uu1 = u1s[f];
        Bv[s][0] = fmaf(wb00, uu0, fmaf(wb01, uu1, bs0));
        Bv[s][1] = fmaf(wb10, uu0, fmaf(wb11, uu1, bs1));
      }
    } else {                         // odd jj: jj=1..15 -> j even 0..14
#pragma unroll
      for (int s = 0; s < 8; ++s) {
        int f = cidx * 16 + (2 * s);
        float uu0 = u0s[f], uu1 = u1s[f];
        Bv[s][0] = fmaf(wb00, uu0, fmaf(wb01, uu1, bs0));
        Bv[s][1] = fmaf(wb10, uu0, fmaf(wb11, uu1, bs1));
      }
    }

    // 12 chained f32 WMMAs.
    v8f acc0 = {0.f, 0.f, 0.f, 0.f, 0.f, 0.f, 0.f, 0.f};
    v8f acc1 = {0.f, 0.f, 0.f, 0.f, 0.f, 0.f, 0.f, 0.f};
#pragma unroll
    for (int s = 0; s < 4; ++s) {
      v2f a0; a0.x = Av0[s][0]; a0.y = Av0[s][1];
      v2f bb; bb.x = Bv[s][0];  bb.y = Bv[s][1];
      acc0 = __builtin_amdgcn_wmma_f32_16x16x4_f32(false, a0, false, bb,
                                                   (short)0, acc0, false, false);
    }
#pragma unroll
    for (int s = 0; s < 8; ++s) {
      v2f a1; a1.x = Av1[s][0]; a1.y = Av1[s][1];
      v2f bb; bb.x = Bv[s][0];  bb.y = Bv[s][1];
      acc1 = __builtin_amdgcn_wmma_f32_16x16x4_f32(false, a1, false, bb,
                                                   (short)0, acc1, false, false);
    }

    // Dump D to LDS: VGPR g -> row m = g + 8p; m = 2*i_local + r.
#pragma unroll
    for (int g = 0; g < 8; ++g) {
      int mm = g + 8 * p;
      sArr[(mm >> 1)][cidx][mm & 1]     = acc0[g];   // i = 0..7
      sArr[8 + (mm >> 1)][cidx][mm & 1] = acc1[g];   // i = 8..15
    }
    __syncthreads();

    // Carry: s_16 = A s_15 + d_15  (next tile's init, and out[15] needs it)
    if (lane < 16) {
      int c = lane;
      float s150 = sArr[15][c][0], s151 = sArr[15][c][1];
      int f = c * 16 + 15;
      float uu0 = u0s[f], uu1 = u1s[f];
      float d0 = fmaf(wb00, uu0, fmaf(wb01, uu1, bs0));
      float d1 = fmaf(wb10, uu0, fmaf(wb11, uu1, bs1));
      sArr[16][c][0] = fmaf(wa00, s150, fmaf(wa01, s151, d0));
      sArr[16][c][1] = fmaf(wa10, s150, fmaf(wa11, s151, d1));
    }
    __syncthreads();

    // Epilogue: out[t] = s_i + (u0*dt)*(cos u1, sin u1) - s_{i+1}; b128 stores.
    int c = lane >> 1;
    int ib = (lane & 1) * 8;
#pragma unroll
    for (int q = 0; q < 8; q += 2) {
      float4 ov;
      {
        int i = ib + q;
        int f = c * 16 + i;
        float cf = u0s[f] * dts[f];
        float sn_, cs_;
        __sincosf(u1s[f], &sn_, &cs_);
        ov.x = fmaf(cf, cs_, sArr[i][c][0]) - sArr[i + 1][c][0];
        ov.y = fmaf(cf, sn_, sArr[i][c][1]) - sArr[i + 1][c][1];
      }
      {
        int i = ib + q + 1;
        int f = c * 16 + i;
        float cf = u0s[f] * dts[f];
        float sn_, cs_;
        __sincosf(u1s[f], &sn_, &cs_);
        ov.z = fmaf(cf, cs_, sArr[i][c][0]) - sArr[i + 1][c][0];
        ov.w = fmaf(cf, sn_, sArr[i][c][1]) - sArr[i + 1][c][1];
      }
      size_t tt = (size_t)(chunkBase + c) * LCH + tbase + ib + q;  // even
      *(float4*)(out2 + tt) = ov;
    }
    __syncthreads();
  }
}

// ---------------------------------------------------------------------------
extern "C" void kernel_launch(void* const* d_in, const int* in_sizes, int n_in,
                              void* d_out, int out_size, void* d_ws, size_t ws_size,
                              hipStream_t stream) {
  (void)in_sizes; (void)n_in; (void)out_size; (void)ws_size;
  const float* x0 = (const float*)d_in[0];
  const float* u  = (const float*)d_in[1];
  const float* td = (const float*)d_in[2];
  const float* WA = (const float*)d_in[3];
  const float* bA = (const float*)d_in[4];
  const float* WB = (const float*)d_in[5];
  const float* bB = (const float*)d_in[6];

  float2* S = (float2*)d_ws;        // NCH chunk sums   (512 KB)
  float2* X = S + NCH;              // NCH chunk states (512 KB)

  pinn_k1_chunksum<<<NCH / 256, 256, 0, stream>>>(u, WA, bA, WB, bB, S);
  pinn_k2_scan<<<1, 256, 0, stream>>>(x0, WA, S, X);
  pinn_k3_wmma<<<NCH / CPW, 32, 0, stream>>>(u, td, WA, bA, WB, bB, X,
                                             (float2*)d_out);
}